// Codebook_55267639165165
// MI455X (gfx1250) — compile-verified
//
#include <hip/hip_runtime.h>
#include <hip/hip_bf16.h>

// Problem constants (from reference): z [16,256,64,64] f32, embedding [1024,256] f32
#define BATCH 16
#define CDIM  256          // D
#define HW    4096         // 64*64
#define NTOT  (BATCH * HW) // 65536 rows
#define KCB   1024         // codebook size

typedef float v2f __attribute__((ext_vector_type(2)));
typedef float v8f __attribute__((ext_vector_type(8)));
typedef int   v4i __attribute__((vector_size(16)));   // matches builtin's pointee type

#if defined(__has_builtin)
#if __has_builtin(__builtin_amdgcn_global_load_async_to_lds_b128)
#define HAVE_ASYNC_LDS 1
#endif
#endif
#ifndef HAVE_ASYNC_LDS
#define HAVE_ASYNC_LDS 0
#endif

// LDS B-tile geometry: 16 "r" rows, each holding two code sub-tiles
// (codes cbase+r and cbase+16+r), 256 floats each, at a 260-float gap
// (=> 1040B: fits ds_load_2addr_b64 offset 130) and a 532-float row stride
// (532 mod 64 == 20 == 4*odd => 16 rows hit 16 distinct bank groups).
#define BSUB   260
#define BROW   532
#define BTILE_FLOATS (16 * BROW)

// ---------------------------------------------------------------------------
// Kernel 1: e_sq[k] = sum_d embedding[k][d]^2   (one wave32 per code)
// ---------------------------------------------------------------------------
__global__ void __launch_bounds__(256) esq_kernel(const float* __restrict__ emb,
                                                  float* __restrict__ esq) {
    int gwave = (blockIdx.x * blockDim.x + threadIdx.x) >> 5;
    int lane  = threadIdx.x & 31;
    if (gwave >= KCB) return;
    const float* row = emb + (size_t)gwave * CDIM + lane * 8;
    float s = 0.0f;
#pragma unroll
    for (int q = 0; q < 8; ++q) { float v = row[q]; s += v * v; }
#pragma unroll
    for (int m = 16; m >= 1; m >>= 1) s += __shfl_xor(s, m, 32);
    if (lane == 0) esq[gwave] = s;
}

// ---------------------------------------------------------------------------
// Kernel 2/5: batched transpose  src[B, I, J] -> dst[B, J, I]
// block (32,8); grid (J/32, I/32, B); 32x33 LDS tile for coalescing
// ---------------------------------------------------------------------------
__global__ void __launch_bounds__(256) transpose_kernel(const float* __restrict__ src,
                                                        float* __restrict__ dst,
                                                        int I, int J) {
    __shared__ float tile[32][33];
    int j0 = blockIdx.x * 32;
    int i0 = blockIdx.y * 32;
    size_t base = (size_t)blockIdx.z * (size_t)I * (size_t)J;
    int tx = threadIdx.x;   // 0..31
    int ty = threadIdx.y;   // 0..7
#pragma unroll
    for (int q = 0; q < 4; ++q) {
        int i = i0 + ty + q * 8;
        tile[ty + q * 8][tx] = src[base + (size_t)i * J + (j0 + tx)];
    }
    __syncthreads();
#pragma unroll
    for (int q = 0; q < 4; ++q) {
        int j = j0 + ty + q * 8;
        dst[base + (size_t)j * I + (i0 + tx)] = tile[tx][ty + q * 8];
    }
}

// ---------------------------------------------------------------------------
// Kernel 3: WMMA distance + argmin.
// One wave32 per 16-row tile of encoded_flat; 8 waves per block share a
// 32-code B stage in LDS (async-copied from global when available).
// Each stage: two 16x16 column tiles computed simultaneously so a single
// A-operand global_load_b64 feeds two v_wmma_f32_16x16x4_f32.
// Running per-row argmin of (e_sq[code] - 2*cross); x^2 is row-constant.
//
// f32 16x16x4 operand layout (ISA 7.12.2):
//   A/B: lane L -> slot M/N = L%16, VGPR0 = K=(2*(L>=16)), VGPR1 = K+1 (float2)
//   C:   VGPR i -> M = i (lanes 0-15) or 8+i (lanes 16-31), N = L%16
// ---------------------------------------------------------------------------
__global__ void __launch_bounds__(256) vq_kernel(const float* __restrict__ enc,
                                                 const float* __restrict__ emb,
                                                 const float* __restrict__ esq,
                                                 float* __restrict__ out_idx_f) {
    __shared__ float btile[BTILE_FLOATS];

    const int tid  = threadIdx.x;
    const int wave = tid >> 5;        // 0..7
    const int lane = tid & 31;
    const int r    = lane & 15;       // row (A) / col (B) slot
    const int hi   = lane >> 4;       // K sub-offset select
    const int n0   = (blockIdx.x * 8 + wave) * 16;

    const float* aPtr = enc + (size_t)(n0 + r) * CDIM + hi * 2;

    // staging decomposition: 256 threads x 32 floats = 32 codes x 256 floats
    const int sc = tid >> 3;          // code-in-stage 0..31
    const int sd = (tid & 7) * 32;    // 32-float span within the code row
    float* sdst = &btile[(sc & 15) * BROW + (sc >> 4) * BSUB + sd];

    float best_d[8];
    int   best_i[8];
#pragma unroll
    for (int i = 0; i < 8; ++i) { best_d[i] = 3.4e38f; best_i[i] = 0; }

    for (int ct = 0; ct < KCB / 32; ++ct) {
        const int cbase = ct * 32;
        __syncthreads(); // previous stage fully consumed before overwrite
        {
            const float* sg = emb + (size_t)(cbase + sc) * CDIM + sd;
#if HAVE_ASYNC_LDS
#pragma unroll
            for (int q = 0; q < 8; ++q) {
                __builtin_amdgcn_global_load_async_to_lds_b128(
                    (__attribute__((address_space(1))) v4i*)(sg + q * 4),
                    (__attribute__((address_space(3))) v4i*)(sdst + q * 4),
                    0, 0);
            }
            asm volatile("s_wait_asynccnt 0x0" ::: "memory");
#else
#pragma unroll
            for (int q = 0; q < 8; ++q) {
                float4 v = ((const float4*)sg)[q];
                sdst[q * 4 + 0] = v.x; sdst[q * 4 + 1] = v.y;
                sdst[q * 4 + 2] = v.z; sdst[q * 4 + 3] = v.w;
            }
#endif
        }
        __syncthreads();

        v8f acc0 = {0.0f, 0.0f, 0.0f, 0.0f, 0.0f, 0.0f, 0.0f, 0.0f};
        v8f acc1 = {0.0f, 0.0f, 0.0f, 0.0f, 0.0f, 0.0f, 0.0f, 0.0f};
        const float* bPtr = &btile[r * BROW + hi * 2];
#pragma unroll 8
        for (int kk = 0; kk < CDIM / 4; ++kk) {
            v2f a  = *(const v2f*)(aPtr + kk * 4);
            v2f b0 = *(const v2f*)(bPtr + kk * 4);
            v2f b1 = *(const v2f*)(bPtr + BSUB + kk * 4);
            acc0 = __builtin_amdgcn_wmma_f32_16x16x4_f32(
                false, a, false, b0, (short)0, acc0, false, false);
            acc1 = __builtin_amdgcn_wmma_f32_16x16x4_f32(
                false, a, false, b1, (short)0, acc1, false, false);
        }

        const int   code0 = cbase + r;
        const int   code1 = cbase + 16 + r;
        const float es0 = esq[code0];
        const float es1 = esq[code1];
#pragma unroll
        for (int i = 0; i < 8; ++i) {
            float d0 = es0 - 2.0f * acc0[i];
            if (d0 < best_d[i]) { best_d[i] = d0; best_i[i] = code0; }
            float d1 = es1 - 2.0f * acc1[i];
            if (d1 < best_d[i]) { best_d[i] = d1; best_i[i] = code1; }
        }
    }

    // Butterfly min-reduce across the 16 lanes of each half (masks 1,2,4,8
    // never cross the 16-lane boundary). Tie-break on smaller index to match
    // argmin's first-occurrence semantics.
#pragma unroll
    for (int m = 8; m >= 1; m >>= 1) {
#pragma unroll
        for (int i = 0; i < 8; ++i) {
            float od = __shfl_xor(best_d[i], m, 32);
            int   oi = __shfl_xor(best_i[i], m, 32);
            if (od < best_d[i] || (od == best_d[i] && oi < best_i[i])) {
                best_d[i] = od; best_i[i] = oi;
            }
        }
    }
    if (r == 0) {
#pragma unroll
        for (int i = 0; i < 8; ++i) {
            int row = hi * 8 + i;       // C layout: lanes 16-31 hold rows 8..15
            out_idx_f[n0 + row] = (float)best_i[i];
        }
    }
}

// ---------------------------------------------------------------------------
// Kernel 4: quantized_flat[n] = enc[n] + (embedding[idx[n]] - enc[n])
// (bit-exact straight-through-estimator expression from the reference)
// ---------------------------------------------------------------------------
__global__ void __launch_bounds__(256) gather_kernel(const float* __restrict__ emb,
                                                     const float* __restrict__ enc,
                                                     const float* __restrict__ idx_f,
                                                     float* __restrict__ qf) {
    int n = blockIdx.x;
    int c = threadIdx.x;
    int k = (int)idx_f[n];
    float e = enc[(size_t)n * CDIM + c];
    float l = emb[(size_t)k * CDIM + c];
    qf[(size_t)n * CDIM + c] = e + (l - e);
}

// ---------------------------------------------------------------------------
// Host launcher
// ---------------------------------------------------------------------------
extern "C" void kernel_launch(void* const* d_in, const int* in_sizes, int n_in,
                              void* d_out, int out_size, void* d_ws, size_t ws_size,
                              hipStream_t stream) {
    const float* z   = (const float*)d_in[0];   // [16,256,64,64]
    const float* emb = (const float*)d_in[1];   // [1024,256]

    float* out  = (float*)d_out;
    float* out0 = out;                              // encoded_flat  [65536,256]
    float* out1 = out + (size_t)NTOT * CDIM;        // quantized_flat[65536,256]
    float* out2 = out1 + (size_t)NTOT * CDIM;       // indices (as float) [65536]
    float* out3 = out2 + NTOT;                      // quantized [16,256,64,64]

    float* esq = (float*)d_ws;                      // 1024 floats

    // 1) e_sq
    esq_kernel<<<dim3((KCB * 32) / 256), dim3(256), 0, stream>>>(emb, esq);

    // 2) z [B,C,HW] -> encoded_flat [B,HW,C]  (output 0)
    transpose_kernel<<<dim3(HW / 32, CDIM / 32, BATCH), dim3(32, 8), 0, stream>>>(
        z, out0, CDIM, HW);

    // 3) WMMA distances + argmin -> output 2
    vq_kernel<<<dim3(NTOT / (16 * 8)), dim3(256), 0, stream>>>(out0, emb, esq, out2);

    // 4) STE gather -> quantized_flat (output 1)
    gather_kernel<<<dim3(NTOT), dim3(CDIM), 0, stream>>>(emb, out0, out2, out1);

    // 5) quantized_flat [B,HW,C] -> quantized [B,C,HW]  (output 3)
    transpose_kernel<<<dim3(CDIM / 32, HW / 32, BATCH), dim3(32, 8), 0, stream>>>(
        out1, out3, HW, CDIM);
}